// SimpleHyperGCN_56384330662517
// MI455X (gfx1250) — compile-verified
//
#include <hip/hip_runtime.h>

typedef __attribute__((ext_vector_type(2))) float v2f;
typedef __attribute__((ext_vector_type(8))) float v8f;

#define N_NODES 100000
#define N_EDGES 25000
#define N_INC   600000
#define DIM     128
#define EDGE_ATTR_DIM 32

// ---------------------------------------------------------------- utilities

__global__ void zero_f32(float* __restrict__ p, int n) {
    int i = blockIdx.x * blockDim.x + threadIdx.x;
    if (i < n) p[i] = 0.0f;
}

__global__ void copy_f32(const float* __restrict__ src, float* __restrict__ dst, int n) {
    int i = blockIdx.x * blockDim.x + threadIdx.x;
    if (i < n) dst[i] = src[i];
}

__global__ void degree_kernel(const int* __restrict__ node_idx,
                              const int* __restrict__ edge_idx,
                              float* __restrict__ deg_n,
                              float* __restrict__ deg_e) {
    int i = blockIdx.x * blockDim.x + threadIdx.x;
    if (i < N_INC) {
        atomicAdd(&deg_n[node_idx[i]], 1.0f);
        atomicAdd(&deg_e[edge_idx[i]], 1.0f);
    }
}

__global__ void invert_kernel(float* __restrict__ p, int n) {
    int i = blockIdx.x * blockDim.x + threadIdx.x;
    if (i < n) {
        float v = p[i];
        p[i] = (v > 0.0f) ? (1.0f / v) : 0.0f;
    }
}

// ---------------------------------------------------------------- WMMA GEMM
// C[100000x128] = A[100000x128] @ W[128x128], fp32, CDNA5 WMMA 16x16x4.
// Block = 256 threads = 8 waves. Block handles 16 rows; wave w handles the
// 16-column tile n0 = 16*w. A strip (16x128 f32 = 8KB) staged in LDS so all
// global reads of the strip finish before the barrier -> safe even if C == A.
__global__ __launch_bounds__(256) void gemm_wmma_f32(
        const float* __restrict__ A,
        const float* __restrict__ W,
        float* __restrict__ C) {
    __shared__ float lds[16 * DIM];

    const int t    = threadIdx.x;
    const int lane = t & 31;
    const int wave = t >> 5;          // 0..7 -> N tile
    const int m0   = blockIdx.x * 16; // row base (100000 = 6250*16 exact)

    // cooperative load of 16x128 A strip, coalesced (2048 floats / 256 thr)
    const float* Abase = A + (size_t)m0 * DIM;
#pragma unroll
    for (int j = 0; j < 8; ++j) {
        int idx = t + j * 256;
        lds[idx] = Abase[idx];
    }
    __syncthreads();

    const int half = lane >> 4;   // 0: lanes 0-15, 1: lanes 16-31
    const int l15  = lane & 15;   // M (for A) / N (for B,C,D) within tile
    const int n0   = wave * 16;

    v8f acc = {};
#pragma unroll
    for (int kk = 0; kk < DIM; kk += 4) {
        const int kb = kk + 2 * half;    // even -> 8B aligned LDS read
        // A fragment: VGPR0 = K={0,2}, VGPR1 = K={1,3} (ISA 16x4 f32 layout)
        v2f a = *(const v2f*)&lds[l15 * DIM + kb];
        // B fragment mirrors A's K split; lanes stripe N
        v2f b;
        b.x = W[(size_t)(kb    ) * DIM + n0 + l15];
        b.y = W[(size_t)(kb + 1) * DIM + n0 + l15];
        acc = __builtin_amdgcn_wmma_f32_16x16x4_f32(
            /*neg_a=*/false, a, /*neg_b=*/false, b,
            /*c_mod=*/(short)0, acc, /*reuse_a=*/false, /*reuse_b=*/false);
    }

    // C/D layout: VGPR r -> M = r + 8*half, N = l15
#pragma unroll
    for (int r = 0; r < 8; ++r) {
        int m = m0 + r + 8 * half;
        C[(size_t)m * DIM + n0 + l15] = acc[r];
    }
}

// ------------------------------------------------------------- scatter ops
// One wave per incidence; lane owns a float4 slice of the 128-dim feature.

__global__ __launch_bounds__(256) void scatter_nodes_to_edges(
        const float* __restrict__ hW,
        const int* __restrict__ node_idx,
        const int* __restrict__ edge_idx,
        float* __restrict__ e_acc) {
    int t    = blockIdx.x * blockDim.x + threadIdx.x;
    int inc  = t >> 5;
    int lane = t & 31;
    if (inc >= N_INC) return;
    int n = node_idx[inc];
    int e = edge_idx[inc];
    const float4 v = *(const float4*)(hW + (size_t)n * DIM + lane * 4);
    float* dst = e_acc + (size_t)e * DIM + lane * 4;
    atomicAdd(dst + 0, v.x);
    atomicAdd(dst + 1, v.y);
    atomicAdd(dst + 2, v.z);
    atomicAdd(dst + 3, v.w);
}

__global__ void scale_edges(float* __restrict__ e_acc,
                            const float* __restrict__ Binv) {
    int i = blockIdx.x * blockDim.x + threadIdx.x;   // over N_EDGES*DIM
    if (i < N_EDGES * DIM) e_acc[i] *= Binv[i >> 7];
}

__global__ __launch_bounds__(256) void scatter_edges_to_nodes(
        const float* __restrict__ e_feat,
        const int* __restrict__ node_idx,
        const int* __restrict__ edge_idx,
        float* __restrict__ n_acc) {
    int t    = blockIdx.x * blockDim.x + threadIdx.x;
    int inc  = t >> 5;
    int lane = t & 31;
    if (inc >= N_INC) return;
    int n = node_idx[inc];
    int e = edge_idx[inc];
    const float4 v = *(const float4*)(e_feat + (size_t)e * DIM + lane * 4);
    float* dst = n_acc + (size_t)n * DIM + lane * 4;
    atomicAdd(dst + 0, v.x);
    atomicAdd(dst + 1, v.y);
    atomicAdd(dst + 2, v.z);
    atomicAdd(dst + 3, v.w);
}

__global__ void finalize_nodes(const float* __restrict__ n_acc,
                               const float* __restrict__ Dinv,
                               const float* __restrict__ bias,
                               float* __restrict__ out,
                               int do_relu) {
    int i = blockIdx.x * blockDim.x + threadIdx.x;   // over N_NODES*DIM
    if (i >= N_NODES * DIM) return;
    int n = i >> 7;
    int d = i & (DIM - 1);
    float v = Dinv[n] * n_acc[i] + bias[d];
    if (do_relu) v = fmaxf(v, 0.0f);
    out[i] = v;
}

// ---------------------------------------------------------------- launcher

extern "C" void kernel_launch(void* const* d_in, const int* in_sizes, int n_in,
                              void* d_out, int out_size, void* d_ws, size_t ws_size,
                              hipStream_t stream) {
    (void)in_sizes; (void)n_in; (void)out_size; (void)ws_size;

    const float* x        = (const float*)d_in[0];
    const int*   node_idx = (const int*)  d_in[1];
    const int*   edge_idx = (const int*)  d_in[2];
    const float* hattr    = (const float*)d_in[3];
    const float* Ws[3]    = { (const float*)d_in[4], (const float*)d_in[6], (const float*)d_in[8] };
    const float* bs[3]    = { (const float*)d_in[5], (const float*)d_in[7], (const float*)d_in[9] };

    float* out = (float*)d_out;                 // [0, 12.8M): h, then hattr

    // Workspace layout (floats):
    float* h_buf = (float*)d_ws;                        // 12,800,000 (h and h@W, ping in place)
    float* e_acc = h_buf + (size_t)N_NODES * DIM;       //  3,200,000
    float* Dinv  = e_acc + (size_t)N_EDGES * DIM;       //    100,000
    float* Binv  = Dinv + N_NODES;                      //     25,000
    float* n_acc = out;                                 // node accumulator aliases output h

    const int B = 256;

    // --- degrees (incidence structure is layer-invariant) ---
    zero_f32<<<(N_NODES + B - 1) / B, B, 0, stream>>>(Dinv, N_NODES);
    zero_f32<<<(N_EDGES + B - 1) / B, B, 0, stream>>>(Binv, N_EDGES);
    degree_kernel<<<(N_INC + B - 1) / B, B, 0, stream>>>(node_idx, edge_idx, Dinv, Binv);
    invert_kernel<<<(N_NODES + B - 1) / B, B, 0, stream>>>(Dinv, N_NODES);
    invert_kernel<<<(N_EDGES + B - 1) / B, B, 0, stream>>>(Binv, N_EDGES);

    const float* cur = x;
    for (int l = 0; l < 3; ++l) {
        // h@W  (in-place safe: LDS-staged A strip per block)
        gemm_wmma_f32<<<N_NODES / 16, 256, 0, stream>>>(cur, Ws[l], h_buf);

        // nodes -> edges
        zero_f32<<<(N_EDGES * DIM + B - 1) / B, B, 0, stream>>>(e_acc, N_EDGES * DIM);
        scatter_nodes_to_edges<<<(N_INC * 32) / B, B, 0, stream>>>(h_buf, node_idx, edge_idx, e_acc);
        scale_edges<<<(N_EDGES * DIM + B - 1) / B, B, 0, stream>>>(e_acc, Binv);

        // edges -> nodes
        zero_f32<<<(N_NODES * DIM + B - 1) / B, B, 0, stream>>>(n_acc, N_NODES * DIM);
        scatter_edges_to_nodes<<<(N_INC * 32) / B, B, 0, stream>>>(e_acc, node_idx, edge_idx, n_acc);

        // D^{-1} * acc + b (+ReLU); layers 0,1 feed h_buf, layer 2 writes d_out in place
        float* dst = (l < 2) ? h_buf : out;
        finalize_nodes<<<(N_NODES * DIM + B - 1) / B, B, 0, stream>>>(n_acc, Dinv, bs[l], dst, l < 2 ? 1 : 0);

        cur = h_buf;
    }

    // second tuple output: hyperedge_attr passthrough
    copy_f32<<<(N_EDGES * EDGE_ATTR_DIM + B - 1) / B, B, 0, stream>>>(
        hattr, out + (size_t)N_NODES * DIM, N_EDGES * EDGE_ATTR_DIM);
}